// RefineDetSimpleLoss_50912542327369
// MI455X (gfx1250) — compile-verified
//
#include <hip/hip_runtime.h>

#define BB 32
#define AA 16320
#define CC 81
#define GG 32
#define NEGINF (-1e30f)

typedef __attribute__((ext_vector_type(2))) float v2f;
typedef __attribute__((ext_vector_type(8))) float v8f;

// ---------- device helpers ----------

__device__ __forceinline__ float iou_fn(float gx1, float gy1, float gx2, float gy2,
                                        float ax1, float ay1, float ax2, float ay2) {
  float tlx = fmaxf(gx1, ax1), tly = fmaxf(gy1, ay1);
  float brx = fminf(gx2, ax2), bry = fminf(gy2, ay2);
  float w = fmaxf(brx - tlx, 0.f), h = fmaxf(bry - tly, 0.f);
  float inter = w * h;
  float areaA = (gx2 - gx1) * (gy2 - gy1);
  float areaB = (ax2 - ax1) * (ay2 - ay1);
  return inter / (areaA + areaB - inter);
}

// anchor in cxcywh; ODM branch refines with decode() inline
__device__ __forceinline__ void get_anchor(const float* __restrict__ anch,
                                           const float* __restrict__ refloc,
                                           int b, int a, int refined,
                                           float& cx, float& cy, float& w, float& h) {
  const float* p = anch + (size_t)a * 4;
  float acx = p[0], acy = p[1], aw = p[2], ah = p[3];
  if (refined) {
    const float* l = refloc + ((size_t)b * AA + a) * 4;
    cx = acx + l[0] * 0.1f * aw;
    cy = acy + l[1] * 0.1f * ah;
    w  = aw * __expf(l[2] * 0.2f);
    h  = ah * __expf(l[3] * 0.2f);
  } else {
    cx = acx; cy = acy; w = aw; h = ah;
  }
}

// monotone float <-> uint mapping (descending float == descending uint)
__device__ __forceinline__ unsigned f2u(float f) {
  unsigned b = __float_as_uint(f);
  return (b & 0x80000000u) ? ~b : (b | 0x80000000u);
}
__device__ __forceinline__ float u2f(unsigned u) {
  unsigned b = (u & 0x80000000u) ? (u & 0x7fffffffu) : ~u;
  return __uint_as_float(b);
}

// ---------- kernels ----------

__global__ void initKernel(float* accum) {
  if (threadIdx.x < 16) accum[threadIdx.x] = 0.f;
}

// per-anchor max/argmax over GT (axis 0 of ov), first-occurrence ties
__global__ void bestTruthKernel(const float* __restrict__ anch, const float* __restrict__ refloc,
                                const float* __restrict__ gtb,
                                float* __restrict__ bov, int* __restrict__ bidx, int refined) {
  __shared__ float g[GG * 4];
  int b = blockIdx.y;
  int a = blockIdx.x * blockDim.x + threadIdx.x;
  if (threadIdx.x < GG * 4) g[threadIdx.x] = gtb[(size_t)b * GG * 4 + threadIdx.x];
  __syncthreads();
  if (a >= AA) return;
  float cx, cy, w, h;
  get_anchor(anch, refloc, b, a, refined, cx, cy, w, h);
  float ax1 = cx - 0.5f * w, ay1 = cy - 0.5f * h, ax2 = cx + 0.5f * w, ay2 = cy + 0.5f * h;
  float best = -1.f; int bi = 0;
  #pragma unroll 4
  for (int gi = 0; gi < GG; ++gi) {
    float v = iou_fn(g[gi * 4], g[gi * 4 + 1], g[gi * 4 + 2], g[gi * 4 + 3], ax1, ay1, ax2, ay2);
    if (v > best) { best = v; bi = gi; }
  }
  bov[(size_t)b * AA + a] = best;
  bidx[(size_t)b * AA + a] = bi;
}

// per-GT argmax over anchors (axis 1), first-occurrence ties; one block per image
__global__ void bestPriorKernel(const float* __restrict__ anch, const float* __restrict__ refloc,
                                const float* __restrict__ gtb, int* __restrict__ bpi, int refined) {
  __shared__ float g[GG * 4];
  __shared__ float rv[256];
  __shared__ int ri[256];
  int b = blockIdx.x, tid = threadIdx.x;
  if (tid < GG * 4) g[tid] = gtb[(size_t)b * GG * 4 + tid];
  __syncthreads();
  for (int gi = 0; gi < GG; ++gi) {
    float gx1 = g[gi * 4], gy1 = g[gi * 4 + 1], gx2 = g[gi * 4 + 2], gy2 = g[gi * 4 + 3];
    float best = -1.f; int bi = 0x7fffffff;
    for (int a = tid; a < AA; a += 256) {
      float cx, cy, w, h;
      get_anchor(anch, refloc, b, a, refined, cx, cy, w, h);
      float v = iou_fn(gx1, gy1, gx2, gy2,
                       cx - 0.5f * w, cy - 0.5f * h, cx + 0.5f * w, cy + 0.5f * h);
      if (v > best) { best = v; bi = a; }   // a ascending => first max kept
    }
    rv[tid] = best; ri[tid] = bi;
    __syncthreads();
    for (int s = 128; s > 0; s >>= 1) {
      if (tid < s) {
        if (rv[tid + s] > rv[tid] || (rv[tid + s] == rv[tid] && ri[tid + s] < ri[tid])) {
          rv[tid] = rv[tid + s]; ri[tid] = ri[tid + s];
        }
      }
      __syncthreads();
    }
    if (tid == 0) bpi[(size_t)b * GG + gi] = ri[0];
    __syncthreads();
  }
}

// guarantee override + conf_t/loc_t encode + per-image positive count; one block per image
__global__ void encodeKernel(const float* __restrict__ anch, const float* __restrict__ refloc,
                             const float* __restrict__ gtb, const int* __restrict__ gtl,
                             const int* __restrict__ bpi, float* __restrict__ bov,
                             int* __restrict__ bidx, int* __restrict__ conft,
                             float* __restrict__ loct, int* __restrict__ posnum,
                             int refined, int zero_labels) {
  __shared__ float g[GG * 4];
  __shared__ int gl[GG];
  __shared__ int pcnt[256];
  int b = blockIdx.x, tid = threadIdx.x;
  if (tid < GG * 4) g[tid] = gtb[(size_t)b * GG * 4 + tid];
  if (tid < GG) gl[tid] = zero_labels ? 0 : gtl[(size_t)b * GG + tid];
  __syncthreads();
  if (tid == 0) {
    // sequential last-wins, matching the reference's .at[].set scatter order
    for (int gi = 0; gi < GG; ++gi) {
      int idx = bpi[(size_t)b * GG + gi];
      bov[(size_t)b * AA + idx] = 2.0f;
      bidx[(size_t)b * AA + idx] = gi;
    }
  }
  __syncthreads();
  int cnt = 0;
  for (int a = tid; a < AA; a += 256) {
    size_t ia = (size_t)b * AA + a;
    float ov = bov[ia];
    int ti = bidx[ia];
    int cf = (ov < 0.5f) ? 0 : (gl[ti] + 1);
    conft[ia] = cf;
    cnt += (cf > 0);
    float gx1 = g[ti * 4], gy1 = g[ti * 4 + 1], gx2 = g[ti * 4 + 2], gy2 = g[ti * 4 + 3];
    float mcx = 0.5f * (gx1 + gx2), mcy = 0.5f * (gy1 + gy2);
    float mw = gx2 - gx1, mh = gy2 - gy1;
    float cx, cy, w, h;
    get_anchor(anch, refloc, b, a, refined, cx, cy, w, h);
    float* o = loct + ia * 4;
    o[0] = (mcx - cx) / (0.1f * w);
    o[1] = (mcy - cy) / (0.1f * h);
    o[2] = __logf(mw / w) * 5.0f;
    o[3] = __logf(mh / h) * 5.0f;
  }
  pcnt[tid] = cnt;
  __syncthreads();
  for (int s = 128; s > 0; s >>= 1) { if (tid < s) pcnt[tid] += pcnt[tid + s]; __syncthreads(); }
  if (tid == 0) posnum[b] = pcnt[0];
}

// one wave32 per row: ce = logsumexp(x) - x[t]; ceneg = (t==0) ? ce : -inf
__global__ void ceKernel(const float* __restrict__ conf, const int* __restrict__ conft,
                         float* __restrict__ ce, float* __restrict__ ceneg,
                         int nc, long rows) {
  int lane = threadIdx.x & 31;
  long wid = ((long)blockIdx.x * blockDim.x + threadIdx.x) >> 5;
  long nw = ((long)gridDim.x * blockDim.x) >> 5;
  for (long r = wid; r < rows; r += nw) {
    const float* x = conf + (size_t)r * nc;
    if (r + nw < rows) __builtin_prefetch(conf + (size_t)(r + nw) * nc + lane, 0, 0);
    int c0 = lane, c1 = lane + 32, c2 = lane + 64;
    float x0 = (c0 < nc) ? x[c0] : 0.f;
    float x1 = (c1 < nc) ? x[c1] : 0.f;
    float x2 = (c2 < nc) ? x[c2] : 0.f;
    float m = -3.4e38f;
    if (c0 < nc) m = fmaxf(m, x0);
    if (c1 < nc) m = fmaxf(m, x1);
    if (c2 < nc) m = fmaxf(m, x2);
    for (int off = 16; off > 0; off >>= 1) m = fmaxf(m, __shfl_xor(m, off, 32));
    float s = 0.f;
    if (c0 < nc) s += __expf(x0 - m);
    if (c1 < nc) s += __expf(x1 - m);
    if (c2 < nc) s += __expf(x2 - m);
    for (int off = 16; off > 0; off >>= 1) s += __shfl_xor(s, off, 32);
    if (lane == 0) {
      int t = conft[r];
      float v = m + __logf(s) - x[t];
      ce[r] = v;
      ceneg[r] = (t == 0) ? v : NEGINF;
    }
  }
}

// WMMA f32 reduction: B = ones(4x16) => D[m,n] = sum_k A[m,k] + C[m,n] (exact f32).
// Accumulates N, sum(pos*ce), sum(pos*smoothL1) — 64 anchors per WMMA triple per wave.
__global__ void wmmaReduceKernel(const float* __restrict__ ce, const int* __restrict__ conft,
                                 const float* __restrict__ ploc, const float* __restrict__ loct,
                                 float* __restrict__ accum, long total) {
  int lane = threadIdx.x & 31;
  long wid = ((long)blockIdx.x * blockDim.x + threadIdx.x) >> 5;
  long nw = ((long)gridDim.x * blockDim.x) >> 5;
  v8f cN = {}, cC = {}, cL = {};
  v2f ones; ones[0] = 1.f; ones[1] = 1.f;
  for (long base = wid * 64; base < total; base += nw * 64) {
    v2f aN, aC, aL;
    #pragma unroll
    for (int half = 0; half < 2; ++half) {
      long i = base + half * 32 + lane;
      float n = 0.f, cv = 0.f, lv = 0.f;
      if (i < total && conft[i] > 0) {
        n = 1.f;
        cv = ce[i];
        const float* pp = ploc + (size_t)i * 4;
        const float* tt = loct + (size_t)i * 4;
        #pragma unroll
        for (int j = 0; j < 4; ++j) {
          float d = pp[j] - tt[j];
          float ad = fabsf(d);
          lv += (ad < 1.f) ? (0.5f * d * d) : (ad - 0.5f);
        }
      }
      aN[half] = n; aC[half] = cv; aL[half] = lv;
    }
    // EXEC is all-ones here (uniform loop bound per wave), as WMMA requires.
    cN = __builtin_amdgcn_wmma_f32_16x16x4_f32(false, aN, false, ones, (short)0, cN, false, false);
    cC = __builtin_amdgcn_wmma_f32_16x16x4_f32(false, aC, false, ones, (short)0, cC, false, false);
    cL = __builtin_amdgcn_wmma_f32_16x16x4_f32(false, aL, false, ones, (short)0, cL, false, false);
  }
  float sN = 0.f, sC = 0.f, sL = 0.f;
  #pragma unroll
  for (int v = 0; v < 8; ++v) { sN += cN[v]; sC += cC[v]; sL += cL[v]; }
  for (int off = 16; off > 0; off >>= 1) {
    sN += __shfl_xor(sN, off, 32);
    sC += __shfl_xor(sC, off, 32);
    sL += __shfl_xor(sL, off, 32);
  }
  if (lane == 0) {
    atomicAdd(&accum[0], sN * (1.f / 16.f));   // each row-sum replicated across 16 cols
    atomicAdd(&accum[1], sC * (1.f / 16.f));
    atomicAdd(&accum[2], sL * (1.f / 16.f));
  }
}

// per-image top-neg_num sum of negative CE via 32-bit radix threshold select; one block per image
__global__ void negSumKernel(const float* __restrict__ ceneg, const int* __restrict__ posnum,
                             float* __restrict__ accum) {
  __shared__ int scnt[256];
  __shared__ float sred[256];
  __shared__ unsigned sthr;
  int b = blockIdx.x, tid = threadIdx.x;
  const float* x = ceneg + (size_t)b * AA;
  int pn = posnum[b];
  int neg = min(pn * 3, AA - pn);
  if (neg < 10) neg = 10;
  unsigned t = 0u;
  for (int bit = 31; bit >= 0; --bit) {
    unsigned cand = t | (1u << bit);
    int c = 0;
    for (int a = tid; a < AA; a += 256) c += (f2u(x[a]) >= cand);
    scnt[tid] = c;
    __syncthreads();
    for (int s = 128; s > 0; s >>= 1) { if (tid < s) scnt[tid] += scnt[tid + s]; __syncthreads(); }
    if (tid == 0) sthr = (scnt[0] >= neg) ? cand : t;
    __syncthreads();
    t = sthr;
    __syncthreads();
  }
  // t == k-th largest sortable key; exact tie handling below
  float sum = 0.f; int cgt = 0;
  for (int a = tid; a < AA; a += 256) {
    float f = x[a];
    unsigned u = f2u(f);
    if (u > t && f > 0.5f * NEGINF) { sum += f; cgt++; }
  }
  sred[tid] = sum; scnt[tid] = cgt;
  __syncthreads();
  for (int s = 128; s > 0; s >>= 1) {
    if (tid < s) { sred[tid] += sred[tid + s]; scnt[tid] += scnt[tid + s]; }
    __syncthreads();
  }
  if (tid == 0) {
    float ft = u2f(t);
    float tot = sred[0];
    if (ft > 0.5f * NEGINF) tot += (float)(neg - scnt[0]) * ft;
    atomicAdd(&accum[3], tot);
  }
}

__global__ void finalizeKernel(const float* __restrict__ accum, float* __restrict__ out) {
  if (threadIdx.x == 0 && blockIdx.x == 0) {
    float Na = accum[0], pa = accum[1], la = accum[2], na = accum[3];
    float No = accum[4], po = accum[5], lo = accum[6], no = accum[7];
    float arm_cls = (pa + na) / Na, arm_loc = la / Na;
    float odm_cls = (po + no) / No, odm_loc = lo / No;
    out[0] = arm_cls + arm_loc + odm_cls + odm_loc;
    out[1] = odm_cls;
    out[2] = odm_loc;
    out[3] = arm_cls;
    out[4] = arm_loc;
  }
}

// ---------- host launcher ----------

extern "C" void kernel_launch(void* const* d_in, const int* in_sizes, int n_in,
                              void* d_out, int out_size, void* d_ws, size_t ws_size,
                              hipStream_t stream) {
  const float* objectness = (const float*)d_in[0];   // [B,A,2]
  const float* refine_loc = (const float*)d_in[1];   // [B,A,4]
  const float* pred_conf  = (const float*)d_in[2];   // [B,A,C]
  const float* pred_loc   = (const float*)d_in[3];   // [B,A,4]
  const float* anchors    = (const float*)d_in[4];   // [1,A,4]
  const float* gt_boxes   = (const float*)d_in[5];   // [B,G,4]
  const int*   gt_labels  = (const int*)d_in[6];     // [B,G]
  float* out = (float*)d_out;

  // workspace carve-up (branches processed sequentially, scratch reused)
  char* w = (char*)d_ws;
  size_t off = 0;
  float* accum = (float*)(w + off); off += 64;
  float* bov   = (float*)(w + off); off += (size_t)BB * AA * 4;
  int*   bidx  = (int*)  (w + off); off += (size_t)BB * AA * 4;
  int*   bpi   = (int*)  (w + off); off += (size_t)BB * GG * 4;
  int*   conft = (int*)  (w + off); off += (size_t)BB * AA * 4;
  float* loct  = (float*)(w + off); off += (size_t)BB * AA * 4 * 4;
  float* ce    = (float*)(w + off); off += (size_t)BB * AA * 4;
  float* ceneg = (float*)(w + off); off += (size_t)BB * AA * 4;
  int*   posn  = (int*)  (w + off); off += (size_t)BB * 4;

  const long rows = (long)BB * AA;

  initKernel<<<1, 32, 0, stream>>>(accum);

  auto run_branch = [&](const float* conf, int nc, const float* ploc,
                        int refined, int zero_labels, float* acc) {
    dim3 gbt((AA + 255) / 256, BB);
    bestTruthKernel<<<gbt, 256, 0, stream>>>(anchors, refine_loc, gt_boxes, bov, bidx, refined);
    bestPriorKernel<<<BB, 256, 0, stream>>>(anchors, refine_loc, gt_boxes, bpi, refined);
    encodeKernel<<<BB, 256, 0, stream>>>(anchors, refine_loc, gt_boxes, gt_labels,
                                         bpi, bov, bidx, conft, loct, posn,
                                         refined, zero_labels);
    ceKernel<<<1024, 256, 0, stream>>>(conf, conft, ce, ceneg, nc, rows);
    wmmaReduceKernel<<<256, 256, 0, stream>>>(ce, conft, ploc, loct, acc, rows);
    negSumKernel<<<BB, 256, 0, stream>>>(ceneg, posn, acc);
  };

  // ARM: binary objectness vs fixed anchors, labels zeroed
  run_branch(objectness, 2, refine_loc, /*refined=*/0, /*zero_labels=*/1, accum + 0);
  // ODM: 81-class conf vs refined anchors, real labels
  run_branch(pred_conf, CC, pred_loc, /*refined=*/1, /*zero_labels=*/0, accum + 4);

  finalizeKernel<<<1, 32, 0, stream>>>(accum, out);
}